// OutliersQLinearUnstruct_75608604279076
// MI455X (gfx1250) — compile-verified
//
#include <hip/hip_runtime.h>

typedef __attribute__((ext_vector_type(2))) float        v2f;
typedef __attribute__((ext_vector_type(4))) float        v4f_t;
typedef __attribute__((ext_vector_type(8))) float        v8f;
typedef __attribute__((ext_vector_type(4))) unsigned int v4u_t;

#define IC       4096
#define OC       11008
#define NB       32
#define ROWS     16            // OC rows per workgroup
#define THREADS  256           // 8 wave32
#define KC       256           // K elements per staged chunk
#define NCHUNK   (IC / KC)     // 16
#define SPC      (KC / 4)      // 64 WMMA K-steps per chunk
#define FSTRIDE  132           // floats per lane in frag buffer (128 + 4 pad -> conflict-free)
#define FRAGF    (32 * FSTRIDE)

// ---------------------------------------------------------------------------
// Kernel 0: swizzle x[32,4096] into WMMA B-fragment order in d_ws (512 KB).
//   xf[(bTile*1024 + S)*32 + lane] = { x[b][k], x[b][k+1] }
//   with b = bTile*16 + lane%16, k = 4*S + (lane/16)*2   (ISA B 4x16 layout)
// ---------------------------------------------------------------------------
__global__ __launch_bounds__(256)
void swizzle_x_kernel(const float* __restrict__ x, float2* __restrict__ xf)
{
    int idx = blockIdx.x * 256 + threadIdx.x;        // 0 .. 65535
    int bT  = idx >> 15;
    int rem = idx & 32767;
    int S   = rem >> 5;
    int l   = rem & 31;
    int b   = bT * 16 + (l & 15);
    int k   = (S << 2) + ((l >> 4) << 1);
    xf[idx] = *(const float2*)(x + (size_t)b * IC + k);
}

// ---------------------------------------------------------------------------
// Kernel 1: main fused kernel. 16 OC rows per WG, ~36 KB LDS -> multiple WGs
// per WGP so HBM streaming (phase A) of some WGs overlaps WMMA of others.
// Weight+mask are read twice but the 2nd read hits the 192MB L2 (tile-hot),
// so HBM traffic stays at the single-pass 225 MB minimum.
// ---------------------------------------------------------------------------
__global__ __launch_bounds__(THREADS)
void OutliersQLinearUnstruct_kernel(const float*         __restrict__ weight,
                                    const float*         __restrict__ bias,
                                    const unsigned char* __restrict__ mask,
                                    const float2*        __restrict__ xf,
                                    float*               __restrict__ out)
{
    __shared__ float fragbuf[2 * FRAGF];   // 33792 B, double-buffered A-fragments
    __shared__ float rowSum[ROWS], rowAbs[ROWS], scaleA[ROWS], muA[ROWS];
    __shared__ float acc[2 * 256];         // 2 batch tiles x 16x16 partial sums

    const int tid     = threadIdx.x;
    const int rowBase = blockIdx.x * ROWS;
    const int rA      = tid >> 4;          // row owned for staging/stats
    const int subA    = tid & 15;          // 16 threads cooperate per row

    if (tid < ROWS) { rowSum[tid] = 0.f; rowAbs[tid] = 0.f; }
    acc[tid] = 0.f; acc[tid + THREADS] = 0.f;
    __syncthreads();

    const float*         wrowg = weight + (size_t)(rowBase + rA) * IC;
    const unsigned char* mrowg = mask   + (size_t)(rowBase + rA) * IC;

    // ---- Phase A: stream tile from HBM once (fills L2), per-row stats ----
    {
        float s = 0.f, a = 0.f;
        for (int it = 0; it < IC / (16 * 8); ++it) {          // 32 iterations
            int c = (it * 16 + subA) * 8;                     // coalesced 512B/row
            float4 w0 = *(const float4*)(wrowg + c);
            float4 w1 = *(const float4*)(wrowg + c + 4);
            unsigned long long mb = *(const unsigned long long*)(mrowg + c);
            float wv[8] = {w0.x, w0.y, w0.z, w0.w, w1.x, w1.y, w1.z, w1.w};
#pragma unroll
            for (int j = 0; j < 8; ++j) {
                float z = (((mb >> (8 * j)) & 0xffull) != 0ull) ? 0.f : wv[j];
                s += z;
                a += fabsf(z);
            }
        }
        atomicAdd(&rowSum[rA], s);                            // ds_add_f32
        atomicAdd(&rowAbs[rA], a);
    }
    __syncthreads();
    if (tid < ROWS) {
        scaleA[tid] = rowAbs[tid] * (1.0f / IC);
        muA[tid]    = rowSum[tid] * (1.0f / IC);
    }
    __syncthreads();

    const float scale_m = scaleA[rA];
    const float mu_m    = muA[rA];

    const int wave  = tid >> 5;
    const int lane  = tid & 31;
    const int bTile = wave >> 2;           // waves 0-3: b 0..15, waves 4-7: b 16..31
    const int kw    = wave & 3;            // K-slice within a chunk

    // stage(): build w_eff for one 16x256 chunk, pre-swizzled into A-fragment
    // order. 2nd read of weight/mask -> L2-hot, marked nontemporal (last use).
    auto stage = [&](int c, int bufSel) {
        float* fb = fragbuf + bufSel * FRAGF;
        const float*         wp = wrowg + c * KC + subA * 16;
        const unsigned char* mp = mrowg + c * KC + subA * 16;
        v4u_t mraw = __builtin_nontemporal_load((const v4u_t*)mp);
#pragma unroll
        for (int g = 0; g < 4; ++g) {
            v4f_t wv = __builtin_nontemporal_load((const v4f_t*)(wp + g * 4));
            unsigned int mword = mraw[g];
#pragma unroll
            for (int part = 0; part < 4; ++part) {
                float w = wv[part];
                float d = w - mu_m;
                float t = copysignf(scale_m, d);
                float q = (d == 0.f) ? 0.f : t;               // jnp.sign(0)=0
                float we = ((mword >> (8 * part)) & 0xffu) ? w : q;
                int sstep = subA * 4 + g;                     // K-group in chunk
                int l     = rA + ((part >> 1) << 4);          // ISA A 16x4 layout
                int j     = part & 1;
                fb[l * FSTRIDE + sstep * 2 + j] = we;
            }
        }
    };

    // compute(): 16 WMMA K-steps per wave per chunk; A from LDS (b128, two
    // steps, conflict-free via FSTRIDE pad), B from pre-swizzled xf (L2).
    auto compute = [&](int c, int bufSel, v8f& C) {
        const float*  fb  = fragbuf + bufSel * FRAGF + lane * FSTRIDE;
        const float2* xfc = xf + ((size_t)(bTile * (IC / 4) + c * SPC)) * 32 + lane;
        const int sBeg = kw * (SPC / 4);                      // 16 steps
#pragma unroll
        for (int s = sBeg; s < sBeg + SPC / 4; s += 2) {
            v4f_t A2 = *(const v4f_t*)(fb + s * 2);           // ds_load_b128
            float2 b0 = xfc[(size_t)s * 32];                  // coalesced b64
            float2 b1 = xfc[(size_t)(s + 1) * 32];
            v2f A0 = {A2[0], A2[1]};
            v2f A1 = {A2[2], A2[3]};
            v2f B0 = {b0.x, b0.y};
            v2f B1 = {b1.x, b1.y};
            C = __builtin_amdgcn_wmma_f32_16x16x4_f32(false, A0, false, B0,
                                                      (short)0, C, false, false);
            C = __builtin_amdgcn_wmma_f32_16x16x4_f32(false, A1, false, B1,
                                                      (short)0, C, false, false);
        }
    };

    // ---- Phase C: double-buffered chunk pipeline, one barrier per chunk ----
    v8f C = {0.f, 0.f, 0.f, 0.f, 0.f, 0.f, 0.f, 0.f};
    stage(0, 0);
    __syncthreads();
    for (int c = 0; c < NCHUNK; ++c) {
        if (c + 1 < NCHUNK) stage(c + 1, (c + 1) & 1);
        compute(c, c & 1, C);
        __syncthreads();
    }

    // ---- reduce 4 K-slice partials per batch tile (ds_add_f32) ----
    {
        float* accT = acc + bTile * 256;
        const int m   = lane & 15;                            // C layout: N = lane%16
        const int Mhi = (lane >> 4) << 3;                     // lanes 16-31: M = r+8
#pragma unroll
        for (int r = 0; r < 8; ++r) {
            atomicAdd(&accT[(r + Mhi) * 16 + m], C[r]);
        }
    }
    __syncthreads();

    // ---- add bias, write y[b, 0, o] ----
    for (int idx = tid; idx < 512; idx += THREADS) {
        int bt  = idx >> 8;
        int rem = idx & 255;
        int M   = rem >> 4;
        int N   = rem & 15;
        int o   = rowBase + M;
        int b   = bt * 16 + N;
        out[(size_t)b * OC + o] = acc[idx] + bias[o];
    }
}

extern "C" void kernel_launch(void* const* d_in, const int* in_sizes, int n_in,
                              void* d_out, int out_size, void* d_ws, size_t ws_size,
                              hipStream_t stream) {
    (void)in_sizes; (void)n_in; (void)out_size; (void)ws_size;
    const float*         x    = (const float*)d_in[0];           // [32,1,4096]
    const float*         w    = (const float*)d_in[1];           // [11008,4096]
    const float*         bias = (const float*)d_in[2];           // [11008]
    const unsigned char* mask = (const unsigned char*)d_in[3];   // [11008,4096] bool
    float* out = (float*)d_out;                                  // [32,1,11008]

    // d_ws usage: 2 * 1024 * 32 * sizeof(float2) = 524288 bytes (x fragments)
    float2* xf = (float2*)d_ws;

    swizzle_x_kernel<<<dim3(256), dim3(256), 0, stream>>>(x, xf);
    OutliersQLinearUnstruct_kernel<<<dim3(OC / ROWS), dim3(THREADS), 0, stream>>>(
        w, bias, mask, xf, out);
}